// CoarsePyramid_41626823033502
// MI455X (gfx1250) — compile-verified
//
#include <hip/hip_runtime.h>

typedef __attribute__((ext_vector_type(2))) float v2f;
typedef __attribute__((ext_vector_type(8))) float v8f;

#define BB 8
#define TT 64
#define PC 512
#define CATC 3572
#define GROUPS 32
#define KC 64            // K-chunk staged in LDS per block

// gfx1250 async global->LDS path (guarded: falls back to plain staging)
#if defined(__has_builtin)
#  if __has_builtin(__builtin_amdgcn_global_load_async_to_lds_b32) && \
      __has_builtin(__builtin_amdgcn_s_wait_asynccnt)
#    define USE_ASYNC_LDS 1
#  endif
#endif
#ifndef USE_ASYNC_LDS
#  define USE_ASYNC_LDS 0
#endif

typedef __attribute__((address_space(1))) int* gas_i32p;
typedef __attribute__((address_space(3))) int* las_i32p;

// ---------------------------------------------------------------------------
// TN GEMM via V_WMMA_F32_16X16X4_F32, LDS-staged B operand.
//   Y[b, yoff+m, n] = sum_k W[m,k] * X[b,k,n] + bias[m]      (N == 64)
// Block = (32,4): wave w owns m-tile blockIdx.x*4+w and computes ALL four
// 16-col n-tiles (4 accumulators share each A fragment). The block stages
// X chunks [kc x 64] into LDS cooperatively (coalesced rows; async-to-LDS
// when available), shared by all 4 waves.
// Fragment mapping (ISA 7.12.2):
//   A(16x4): lane -> M=lane%16, vgpr v -> K = v + (lane/16)*2
//   B(4x16): lane -> N=lane%16, vgpr r -> K = r + (lane/16)*2
//   C/D:     lane -> N=lane%16, vgpr r -> M = r + (lane/16)*8
// ---------------------------------------------------------------------------
__global__ void gemm_tn_wmma(const float* __restrict__ W,
                             const float* __restrict__ X,
                             const float* __restrict__ bias,
                             float* __restrict__ Y,
                             int M, int K, long xbs, long ybs, int yoff)
{
    __shared__ float xs[KC * 65];              // 64-float rows padded to 65
    const int lane = threadIdx.x;
    const int half = lane >> 4;
    const int l16  = lane & 15;
    const int tid  = threadIdx.y * 32 + lane;
    const int m_base = (blockIdx.x * 4 + threadIdx.y) * 16;
    const int b      = blockIdx.z;

    const float* Xb   = X + (long)b * xbs;
    const float* wrow = W + (long)(m_base + l16) * K + half * 2;
    __builtin_prefetch(wrow, 0, 0);

    v8f acc[4] = {};
    for (int k0 = 0; k0 < K; k0 += KC) {
        const int kc = min(KC, K - k0);        // tail (e.g. K=3572): multiple of 4
        __syncthreads();                       // previous chunk fully consumed
        for (int i = tid; i < kc * 64; i += 128) {
            const int kr = i >> 6, col = i & 63;
            const float* src = Xb + (long)(k0 + kr) * 64 + col;
#if USE_ASYNC_LDS
            __builtin_amdgcn_global_load_async_to_lds_b32(
                (gas_i32p)(float*)src,
                (las_i32p)&xs[kr * 65 + col],
                0, 0);
#else
            xs[kr * 65 + col] = *src;
#endif
        }
#if USE_ASYNC_LDS
        __builtin_amdgcn_s_wait_asynccnt(0);
#endif
        __syncthreads();

        for (int kk = 0; kk < kc; kk += 4) {
            v2f a;
            a.x = wrow[k0 + kk];
            a.y = wrow[k0 + kk + 1];
            const int krow = kk + half * 2;
            const float* xr0 = &xs[krow * 65 + l16];
            const float* xr1 = &xs[(krow + 1) * 65 + l16];
            #pragma unroll
            for (int nt = 0; nt < 4; ++nt) {
                v2f bb;
                bb.x = xr0[nt * 16];
                bb.y = xr1[nt * 16];
                acc[nt] = __builtin_amdgcn_wmma_f32_16x16x4_f32(
                    false, a, false, bb, (short)0, acc[nt], false, false);
            }
        }
    }

    float* Yb = Y + (long)b * ybs + (long)yoff * 64;
    #pragma unroll
    for (int r = 0; r < 8; ++r) {
        const int m = m_base + r + half * 8;
        const float bv = bias[m];
        #pragma unroll
        for (int nt = 0; nt < 4; ++nt)
            Yb[(long)m * 64 + nt * 16 + l16] = acc[nt][r] + bv;
    }
}

// ---------------------------------------------------------------------------
// Anchor-gather conv as a TN GEMM with on-the-fly gathered B matrix.
//   Y[b, yoff+m, t] = sum_{c,k} W[m, c*KK+k] * F[b, c, idx(b,t,k)] * valid(b,t)
// Per-lane t is fixed inside the B fragment, so the KK gather indices live
// in registers (half-selected with compile-time indexing -> no scratch).
// ---------------------------------------------------------------------------
template<int KK>
__global__ void gemm_gather_wmma(const float* __restrict__ W,   // [M, C*KK]
                                 const float* __restrict__ F,   // [B, C, L]
                                 const int*   __restrict__ anc, // [B, T, 2]
                                 const float* __restrict__ bias,
                                 float* __restrict__ Y,
                                 int M, int C, int L,
                                 long fbs, long ybs, int yoff)
{
    const int N = TT;
    const int lane = threadIdx.x;
    const int half = lane >> 4;
    const int l16  = lane & 15;
    const int m_base = (blockIdx.x * blockDim.y + threadIdx.y) * 16;
    const int n_base = blockIdx.y * 16;
    const int b      = blockIdx.z;
    const int t      = n_base + l16;

    const int* a2 = anc + ((long)b * N + t) * 2;
    int left  = min(max(a2[0], 0), L);
    int right = min(max(a2[1], 0), L);
    const float validf = (right > left) ? 1.0f : 0.0f;
    const int seglen = right + 1 - left;

    int idx[KK];
    #pragma unroll
    for (int k = 0; k < KK; ++k)
        idx[k] = min(left + (k * seglen) / KK, L - 1);

    // This lane's two B K-slots per 4-wide step: kk = 4q + half*2 + {0,1}.
    int jdx0[KK / 4], jdx1[KK / 4];
    #pragma unroll
    for (int q = 0; q < KK / 4; ++q) {
        jdx0[q] = half ? idx[4 * q + 2] : idx[4 * q + 0];
        jdx1[q] = half ? idx[4 * q + 3] : idx[4 * q + 1];
    }

    const int Ktot = C * KK;
    const float* wrow = W + (long)(m_base + l16) * Ktot + half * 2;
    const float* Fb   = F + (long)b * fbs;
    __builtin_prefetch(wrow, 0, 0);

    v8f acc = {};
    for (int c = 0; c < C; ++c) {
        const float* frow = Fb + (long)c * L;
        const float* wc   = wrow + c * KK;
        #pragma unroll
        for (int q = 0; q < KK / 4; ++q) {
            v2f a;  a.x  = wc[4 * q];              a.y  = wc[4 * q + 1];
            v2f bb; bb.x = frow[jdx0[q]] * validf; bb.y = frow[jdx1[q]] * validf;
            acc = __builtin_amdgcn_wmma_f32_16x16x4_f32(
                false, a, false, bb, (short)0, acc, false, false);
        }
    }

    float* Yb = Y + (long)b * ybs;
    #pragma unroll
    for (int r = 0; r < 8; ++r) {
        const int m = m_base + r + half * 8;
        Yb[(long)(yoff + m) * N + n_base + l16] = acc[r] + bias[m];
    }
}

// ---------------------------------------------------------------------------
// Boundary max pooling: first C/2 channels pool over segs[...,0:2],
// second C/2 over segs[...,2:4] (inclusive, clamped; s<=e after clamp).
// ---------------------------------------------------------------------------
__global__ void bmp_kernel(const float* __restrict__ F, const int* __restrict__ segs,
                           float* __restrict__ Y, int C, int L,
                           long fbs, long ybs, int yoff)
{
    long tid = (long)blockIdx.x * blockDim.x + threadIdx.x;
    long total = (long)BB * C * TT;
    if (tid >= total) return;
    int t  = (int)(tid % TT);
    int ch = (int)((tid / TT) % C);
    int b  = (int)(tid / ((long)TT * C));

    const int* sg = segs + ((long)b * TT + t) * 4;
    int s, e;
    if (ch < C / 2) { s = sg[0]; e = sg[1]; } else { s = sg[2]; e = sg[3]; }
    s = min(max(s, 0), L - 1);
    e = max(min(max(e, 0), L - 1), s);

    const float* fr = F + (long)b * fbs + (long)ch * L;
    float m = fr[s];
    for (int l = s + 1; l <= e; ++l) m = fmaxf(m, fr[l]);
    Y[(long)b * ybs + (long)(yoff + ch) * TT + t] = m;
}

// ---------------------------------------------------------------------------
// In-place GroupNorm(32) + affine + ReLU on a [C, T] slice of a batch-strided
// buffer. Block per (b, group).
// ---------------------------------------------------------------------------
__global__ void gn_relu_kernel(float* __restrict__ Y,
                               const float* __restrict__ gamma,
                               const float* __restrict__ beta,
                               int C, long ybs, int yoff)
{
    const int b = blockIdx.x, g = blockIdx.y;
    const int cpg = C / GROUPS;
    const long n = (long)cpg * TT;
    float* base = Y + (long)b * ybs + (long)(yoff + g * cpg) * TT;

    __shared__ float ssum[256], ssq[256];
    float s = 0.f, q = 0.f;
    for (long i = threadIdx.x; i < n; i += blockDim.x) {
        float v = base[i]; s += v; q += v * v;
    }
    ssum[threadIdx.x] = s; ssq[threadIdx.x] = q;
    __syncthreads();
    for (int o = blockDim.x / 2; o > 0; o >>= 1) {
        if ((int)threadIdx.x < o) {
            ssum[threadIdx.x] += ssum[threadIdx.x + o];
            ssq[threadIdx.x]  += ssq[threadIdx.x + o];
        }
        __syncthreads();
    }
    const float mean = ssum[0] / (float)n;
    const float var  = ssq[0] / (float)n - mean * mean;
    const float inv  = rsqrtf(var + 1e-5f);
    for (long i = threadIdx.x; i < n; i += blockDim.x) {
        int ch = g * cpg + (int)(i / TT);
        float v = (base[i] - mean) * inv * gamma[ch] + beta[ch];
        base[i] = fmaxf(v, 0.f);
    }
}

// Copy [B,C,T] src into a channel slice of the batch-strided dst.
__global__ void copy_slice(const float* __restrict__ src, float* __restrict__ dst,
                           int C, long sbs, long dbs, int doff)
{
    long tid = (long)blockIdx.x * blockDim.x + threadIdx.x;
    long total = (long)BB * C * TT;
    if (tid >= total) return;
    int t  = (int)(tid % TT);
    int ch = (int)((tid / TT) % C);
    int b  = (int)(tid / ((long)TT * C));
    dst[(long)b * dbs + (long)(doff + ch) * TT + t] =
        src[(long)b * sbs + (long)ch * TT + t];
}

// ---------------------------------------------------------------------------
extern "C" void kernel_launch(void* const* d_in, const int* in_sizes, int n_in,
                              void* d_out, int out_size, void* d_ws, size_t ws_size,
                              hipStream_t stream)
{
    const float* feature    = (const float*)d_in[0];   // [8,512,64]
    const float* flf        = (const float*)d_in[1];   // [8,512,256]
    const int*   segments   = (const int*)  d_in[2];   // [8,64,4]
    const int*   fsegments  = (const int*)  d_in[3];   // [8,64,4]
    const float* lastclass  = (const float*)d_in[4];   // [8,498,64]
    const int*   anchor     = (const int*)  d_in[5];   // [8,64,2]
    const int*   fanchor    = (const int*)  d_in[6];   // [8,64,2]
    /* d_in[7] = order (==0, so K1=16, K0=8) */
    const float* locres     = (const float*)d_in[8];   // [8,2,64]
    const float* w_cur  = (const float*)d_in[9];
    const float* b_cur  = (const float*)d_in[10];
    const float* g_cur  = (const float*)d_in[11];
    const float* be_cur = (const float*)d_in[12];
    const float* w_lr   = (const float*)d_in[13];
    const float* b_lr   = (const float*)d_in[14];
    const float* g_lr   = (const float*)d_in[15];
    const float* be_lr  = (const float*)d_in[16];
    const float* w_roi  = (const float*)d_in[17];
    const float* b_roi  = (const float*)d_in[18];
    const float* g_roi  = (const float*)d_in[19];
    const float* be_roi = (const float*)d_in[20];
    const float* w_cp   = (const float*)d_in[21];      // [512,1024,16] == [512,16384]
    const float* b_cp   = (const float*)d_in[22];
    const float* w_cp0  = (const float*)d_in[23];      // [512,512,8]  == [512,4096]
    const float* b_cp0  = (const float*)d_in[24];
    const float* w_prop = (const float*)d_in[25];      // [512,3572]
    const float* b_prop = (const float*)d_in[26];
    const float* g_prop = (const float*)d_in[27];
    const float* be_prop= (const float*)d_in[28];

    float* out1 = (float*)d_out;                       // [8,512,64]
    float* flr  = (float*)d_out + 262144;              // [8,1024,64]

    const long pcbs  = (long)PC * TT;        // 32768
    const long lrbs  = (long)2 * PC * TT;    // 65536
    const long catbs = (long)CATC * TT;      // 228608
    const long flfbs = (long)PC * 256;       // 131072

    float* cat  = (float*)d_ws;              // [8,3572,64]  ~7.3 MB
    float* bmpf = cat  + (long)BB * catbs;   // [8,512,64]
    float* fcon = bmpf + (long)BB * pcbs;    // [8,512,64]

    const dim3 gblk(32, 4);
    const dim3 gn_grid(BB, GROUPS);

    // 1) feat_lr pre-act -> d_out[262144:]   (M=1024,K=512)
    gemm_tn_wmma<<<dim3(16, 1, BB), gblk, 0, stream>>>(
        w_lr, feature, b_lr, flr, 1024, 512, pcbs, lrbs, 0);
    // 2) fm_short pre-act -> cat[1536:2048]
    gemm_tn_wmma<<<dim3(8, 1, BB), gblk, 0, stream>>>(
        w_cur, feature, b_cur, cat, 512, 512, pcbs, catbs, 1536);
    // 3) BMP(frame_level_feature) -> ws
    bmp_kernel<<<(BB * 512 * TT + 255) / 256, 256, 0, stream>>>(
        flf, fsegments, bmpf, 512, 256, flfbs, pcbs, 0);
    // 4) frame_con = gather-conv(frame_level_feature, K0=8) -> ws
    gemm_gather_wmma<8><<<dim3(8, 4, BB), gblk, 0, stream>>>(
        w_cp0, flf, fanchor, b_cp0, fcon, 512, 512, 256, flfbs, pcbs, 0);
    // 5) GN+ReLU feat_lr (in place, part 2 of d_out)
    gn_relu_kernel<<<gn_grid, 256, 0, stream>>>(flr, g_lr, be_lr, 1024, lrbs, 0);
    // 6) GN+ReLU fm_short (in place, cat slice)
    gn_relu_kernel<<<gn_grid, 256, 0, stream>>>(cat, g_cur, be_cur, 512, catbs, 1536);
    // 7) prop_feature = BMP(feat_lr) -> cat[512:1536]
    bmp_kernel<<<(BB * 1024 * TT + 255) / 256, 256, 0, stream>>>(
        flr, segments, cat, 1024, TT, lrbs, catbs, 512);
    // 8) con_feature = gather-conv(feat_lr, K1=16) -> cat[2546:3058]
    gemm_gather_wmma<16><<<dim3(8, 4, BB), gblk, 0, stream>>>(
        w_cp, flr, anchor, b_cp, cat, 512, 1024, TT, lrbs, catbs, 2546);
    // 9) prop_roi pre-act = w_roi x BMP_frame -> cat[0:512]; 10) GN+ReLU
    gemm_tn_wmma<<<dim3(8, 1, BB), gblk, 0, stream>>>(
        w_roi, bmpf, b_roi, cat, 512, 512, pcbs, catbs, 0);
    gn_relu_kernel<<<gn_grid, 256, 0, stream>>>(cat, g_roi, be_roi, 512, catbs, 0);
    // 11) prop_con pre-act = w_roi x frame_con -> cat[3058:3570]; 12) GN+ReLU
    gemm_tn_wmma<<<dim3(8, 1, BB), gblk, 0, stream>>>(
        w_roi, fcon, b_roi, cat, 512, 512, pcbs, catbs, 3058);
    gn_relu_kernel<<<gn_grid, 256, 0, stream>>>(cat, g_roi, be_roi, 512, catbs, 3058);
    // 13/14) copies: feature_lastclass -> cat[2048:2546]; LocResult -> cat[3570:3572]
    copy_slice<<<(BB * 498 * TT + 255) / 256, 256, 0, stream>>>(
        lastclass, cat, 498, (long)498 * TT, catbs, 2048);
    copy_slice<<<(BB * 2 * TT + 255) / 256, 256, 0, stream>>>(
        locres, cat, 2, (long)2 * TT, catbs, 3570);
    // 15) final conv (K=3572) -> d_out[0:262144]; 16) GN+ReLU in place
    gemm_tn_wmma<<<dim3(8, 1, BB), gblk, 0, stream>>>(
        w_prop, cat, b_prop, out1, 512, 3572, catbs, pcbs, 0);
    gn_relu_kernel<<<gn_grid, 256, 0, stream>>>(out1, g_prop, be_prop, 512, pcbs, 0);
}